// GraphAttentionLayer_27986006901074
// MI455X (gfx1250) — compile-verified
//
#include <hip/hip_runtime.h>

typedef __attribute__((ext_vector_type(2))) float v2f;
typedef __attribute__((ext_vector_type(8))) float v8f;

#define N_NODES 4096
#define F_DIM   512
#define H_HEADS 8
#define D_DIM   64
#define NEG_INF_V (-1e10f)
#define LEAKY   0.2f

// ---------------------------------------------------------------------------
// Stage 1: feats[h,n,d] = sum_f X[n,f] * W[h,f,d]
// One wave per 16x16 output tile, K stepped by 4 via V_WMMA_F32_16X16X4_F32.
// ---------------------------------------------------------------------------
__global__ __launch_bounds__(256) void feats_gemm(const float* __restrict__ X,
                                                  const float* __restrict__ W,
                                                  float* __restrict__ feats) {
    const int lane = threadIdx.x & 31;
    const int wave = threadIdx.x >> 5;
    const int gw   = blockIdx.x * 8 + wave;      // 0..8191
    const int h    = gw >> 10;                   // 1024 tiles per head
    const int t    = gw & 1023;
    const int i0   = (t >> 2) << 4;              // 256 row tiles
    const int d0   = (t & 3) << 4;               // 4 col tiles

    const int m  = lane & 15;                    // A-row / B-col owned by lane
    const int kb = (lane >> 4) << 1;             // K sub-offset {0,2}

    const float* Xrow = X + (size_t)(i0 + m) * F_DIM;
    const float* Wh   = W + (size_t)h * F_DIM * D_DIM + d0 + m;

    v8f acc = {};
    for (int k = 0; k < F_DIM; k += 4) {
        v2f a, b;
        a.x = Xrow[k + kb];
        a.y = Xrow[k + kb + 1];
        b.x = Wh[(size_t)(k + kb) * D_DIM];
        b.y = Wh[(size_t)(k + kb + 1) * D_DIM];
        acc = __builtin_amdgcn_wmma_f32_16x16x4_f32(false, a, false, b,
                                                    (short)0, acc, false, false);
    }

    // C/D layout: VGPR v holds row (v + 8*(lane>>4)), col (lane&15)
    const int rowbase = (lane >> 4) << 3;
    float* outp = feats + (size_t)h * N_NODES * D_DIM + d0 + m;
#pragma unroll
    for (int v = 0; v < 8; ++v) {
        outp[(size_t)(i0 + rowbase + v) * D_DIM] = acc[v];
    }
}

// ---------------------------------------------------------------------------
// Stage 2: e_self[h,n] = feats[h,n,:].a_self[h,:], same for e_neigh
// ---------------------------------------------------------------------------
__global__ __launch_bounds__(256) void scores_kernel(const float* __restrict__ feats,
                                                     const float* __restrict__ a_self,
                                                     const float* __restrict__ a_neigh,
                                                     float* __restrict__ e_self,
                                                     float* __restrict__ e_neigh) {
    const int idx = blockIdx.x * blockDim.x + threadIdx.x;   // 0..32767
    const int h   = idx >> 12;
    const float* f  = feats + (size_t)idx * D_DIM;
    const float* as = a_self + h * D_DIM;
    const float* an = a_neigh + h * D_DIM;
    float s0 = 0.f, s1 = 0.f;
#pragma unroll 8
    for (int d = 0; d < D_DIM; ++d) {
        const float fv = f[d];
        s0 += fv * as[d];
        s1 += fv * an[d];
    }
    e_self[idx]  = s0;
    e_neigh[idx] = s1;
}

// ---------------------------------------------------------------------------
// Stage 3: masked softmax + attn @ feats + bias, relu, concat-heads store.
// One wave per (head, 16-row tile). Two passes over j (max, then exp+WMMA).
// ---------------------------------------------------------------------------
__global__ __launch_bounds__(256) void gat_attn(const float* __restrict__ A,
                                                const float* __restrict__ feats,
                                                const float* __restrict__ e_self,
                                                const float* __restrict__ e_neigh,
                                                const float* __restrict__ bias,
                                                float* __restrict__ out) {
    __shared__ float P[8][16][18];     // exp(score) tile per wave, padded stride
    __shared__ float Lsum[8][16];      // per-row softmax denominators

    const int lane = threadIdx.x & 31;
    const int wave = threadIdx.x >> 5;
    const int gw   = blockIdx.x * 8 + wave;      // 0..2047
    const int h    = gw >> 8;
    const int i0   = (gw & 255) << 4;

    const int m  = lane & 15;                    // row owned for scoring
    const int jg = lane >> 4;                    // j half: 0 -> j 0..7, 1 -> j 8..15
    const int kb = jg << 1;                      // WMMA K sub-offset {0,2}

    const float* Arow = A + (size_t)(i0 + m) * N_NODES + 8 * jg;
    const float* en   = e_neigh + h * N_NODES + 8 * jg;
    const float  es   = e_self[h * N_NODES + i0 + m];

    // ---- pass 1: row max of masked leaky scores ----
    float mx = -3.0e38f;
    for (int j0 = 0; j0 < N_NODES; j0 += 16) {
#pragma unroll
        for (int t = 0; t < 8; ++t) {
            float s = es + en[j0 + t];
            s = (s >= 0.f) ? s : (LEAKY * s);
            s += NEG_INF_V * (1.0f - Arow[j0 + t]);
            mx = fmaxf(mx, s);
        }
    }
    mx = fmaxf(mx, __shfl_xor(mx, 16, 32));

    // ---- pass 2: p = exp(s - mx); acc += P @ feats_tile via WMMA ----
    v8f acc[4] = {v8f{}, v8f{}, v8f{}, v8f{}};
    float lsum = 0.f;
    const float* fb = feats + (size_t)h * N_NODES * D_DIM;

    for (int j0 = 0; j0 < N_NODES; j0 += 16) {
#pragma unroll
        for (int t = 0; t < 8; ++t) {
            float s = es + en[j0 + t];
            s = (s >= 0.f) ? s : (LEAKY * s);
            s += NEG_INF_V * (1.0f - Arow[j0 + t]);
            const float p = __expf(s - mx);
            lsum += p;
            P[wave][m][8 * jg + t] = p;
        }
        __syncthreads();   // all 8 waves have identical trip counts

        // A-fragments of P: lane holds P[m][4*kc + kb + {0,1}]
        v2f afrag[4];
#pragma unroll
        for (int kc = 0; kc < 4; ++kc) {
            afrag[kc].x = P[wave][m][4 * kc + kb];
            afrag[kc].y = P[wave][m][4 * kc + kb + 1];
        }

        const float* fj = fb + (size_t)j0 * D_DIM + m;   // B col n = m
#pragma unroll
        for (int db = 0; db < 4; ++db) {
#pragma unroll
            for (int kc = 0; kc < 4; ++kc) {
                const float* fp = fj + (size_t)(4 * kc + kb) * D_DIM + db * 16;
                v2f b;
                b.x = fp[0];
                b.y = fp[D_DIM];
                acc[db] = __builtin_amdgcn_wmma_f32_16x16x4_f32(
                    false, afrag[kc], false, b, (short)0, acc[db], false, false);
            }
        }
        __syncthreads();   // WAR guard before next tile's P writes
    }

    // ---- epilogue: divide by row sum, add bias, relu, store [N, H*D] ----
    lsum += __shfl_xor(lsum, 16, 32);
    Lsum[wave][m] = lsum;              // both half-groups write identical value
    __syncthreads();

    const int rowbase = jg << 3;       // rows (rowbase+v) held by this lane
    const float* bh = bias + h * D_DIM;
#pragma unroll
    for (int db = 0; db < 4; ++db) {
        const float bv = bh[db * 16 + m];
#pragma unroll
        for (int v = 0; v < 8; ++v) {
            const float l  = Lsum[wave][rowbase + v];
            float val = acc[db][v] / l + bv;
            val = (val > 0.f) ? val : 0.f;
            out[(size_t)(i0 + rowbase + v) * (H_HEADS * D_DIM)
                + h * D_DIM + db * 16 + m] = val;
        }
    }
}

// ---------------------------------------------------------------------------
extern "C" void kernel_launch(void* const* d_in, const int* in_sizes, int n_in,
                              void* d_out, int out_size, void* d_ws, size_t ws_size,
                              hipStream_t stream) {
    const float* X       = (const float*)d_in[0];
    const float* A       = (const float*)d_in[1];
    const float* W       = (const float*)d_in[2];
    const float* b       = (const float*)d_in[3];
    const float* a_self  = (const float*)d_in[4];
    const float* a_neigh = (const float*)d_in[5];
    float* out = (float*)d_out;

    float* feats   = (float*)d_ws;                                   // 8 MB
    float* e_self  = feats + (size_t)H_HEADS * N_NODES * D_DIM;      // 128 KB
    float* e_neigh = e_self + (size_t)H_HEADS * N_NODES;             // 128 KB

    // Stage 1: 8192 tiles, 8 waves (256 thr) per block -> 1024 blocks
    feats_gemm<<<1024, 256, 0, stream>>>(X, W, feats);
    // Stage 2: 32768 dot products
    scores_kernel<<<128, 256, 0, stream>>>(feats, a_self, a_neigh, e_self, e_neigh);
    // Stage 3: 2048 (head,row-tile) waves -> 256 blocks of 8 waves
    gat_attn<<<256, 256, 0, stream>>>(A, feats, e_self, e_neigh, b, out);
}